// WeightedMSELoss_43585328119833
// MI455X (gfx1250) — compile-verified
//
#include <hip/hip_runtime.h>

typedef __attribute__((ext_vector_type(2))) float v2f;
typedef __attribute__((ext_vector_type(4))) float v4f;
typedef __attribute__((ext_vector_type(8))) float v8f;

#define B_DIM 32
#define T_DIM 12
#define H_DIM 448
#define W_DIM 304
#define BT    (B_DIM * T_DIM)        // 384
#define HW    (H_DIM * W_DIM)        // 136192
#define HW4   (HW / 4)               // 34048 (HW divisible by 4)
#define N4    (BT * HW4)             // 13,074,432 float4 elements
#define N_TOT ((double)BT * (double)HW)

__global__ void wmse_zero_acc(double* acc) {
    if (threadIdx.x == 0) *acc = 0.0;
}

__global__ __launch_bounds__(256) void wmse_reduce_kernel(
    const float* __restrict__ pred,
    const float* __restrict__ targ,
    const int*   __restrict__ months,     // [BT], values 1..12
    const float* __restrict__ mweights,   // [12]
    const float* __restrict__ aweights,   // [HW]
    double* __restrict__ acc)
{
    __shared__ float mw_lds[BT];
    __shared__ float wave_sums[8];

    // Preload per-(b,t) month weight into LDS (384 gathered floats per block).
    for (int i = threadIdx.x; i < BT; i += blockDim.x) {
        mw_lds[i] = mweights[months[i] - 1];
    }
    __syncthreads();

    const v4f* __restrict__ p4 = (const v4f*)pred;
    const v4f* __restrict__ t4 = (const v4f*)targ;
    const v4f* __restrict__ a4 = (const v4f*)aweights;

    float lacc = 0.0f;
    const int stride = gridDim.x * blockDim.x;
    for (int g = blockIdx.x * blockDim.x + threadIdx.x; g < N4; g += stride) {
        unsigned bt = (unsigned)g / (unsigned)HW4;          // mul-shift by compiler
        unsigned r  = (unsigned)g - bt * (unsigned)HW4;
        // Non-temporal: pred/targ are streamed exactly once (418 MB);
        // keeps the 545 KB area_weights tile hot in the 192 MB L2.
        v4f p = __builtin_nontemporal_load(&p4[g]);
        v4f t = __builtin_nontemporal_load(&t4[g]);
        v4f a = a4[r];                                      // L2-resident reuse
        float w = mw_lds[bt];
        v4f d = (t - p) * a;
        float ss = d.x * d.x + d.y * d.y + d.z * d.z + d.w * d.w;
        lacc += (w * w) * ss;
    }

    // ---- Wave(32) reduction via V_WMMA_F32_16X16X4_F32 with ones-B ----
    // A (16x4 f32, 2 VGPRs/lane): A.x -> (M=lane%16, K = lane<16 ? 0 : 2),
    // A.y -> K=1 or 3 (set to 0). B = all ones => D[m][n] = acc(m)+acc(m+16).
    // Lane n (0..15): D VGPR r = rowsum(M=r); lanes 16..31: rowsum(M=8+r).
    // Sum of my 8 D regs + shfl_xor(16) partner = full 32-lane sum.
    v2f Amat; Amat.x = lacc; Amat.y = 0.0f;
    v2f Bmat; Bmat.x = 1.0f; Bmat.y = 1.0f;
    v8f Cmat = {};
    Cmat = __builtin_amdgcn_wmma_f32_16x16x4_f32(
        /*neg_a=*/false, Amat, /*neg_b=*/false, Bmat,
        /*c_mod=*/(short)0, Cmat, /*reuse_a=*/false, /*reuse_b=*/false);
    float s = Cmat[0] + Cmat[1] + Cmat[2] + Cmat[3] +
              Cmat[4] + Cmat[5] + Cmat[6] + Cmat[7];
    s += __shfl_xor(s, 16, 32);

    const int lane = threadIdx.x & 31;
    const int wid  = threadIdx.x >> 5;
    if (lane == 0) wave_sums[wid] = s;
    __syncthreads();
    if (threadIdx.x == 0) {
        float bs = 0.0f;
        const int nw = blockDim.x >> 5;
        for (int i = 0; i < nw; ++i) bs += wave_sums[i];
        atomicAdd(acc, (double)bs);   // global_atomic_add_f64
    }
}

__global__ void wmse_finalize(const double* __restrict__ acc,
                              float* __restrict__ out) {
    if (threadIdx.x == 0) out[0] = (float)(*acc / N_TOT);
}

extern "C" void kernel_launch(void* const* d_in, const int* in_sizes, int n_in,
                              void* d_out, int out_size, void* d_ws, size_t ws_size,
                              hipStream_t stream) {
    const float* pred    = (const float*)d_in[0];
    const float* targ    = (const float*)d_in[1];
    const int*   months  = (const int*)d_in[2];
    const float* mw      = (const float*)d_in[3];
    const float* aw      = (const float*)d_in[4];
    double* acc = (double*)d_ws;          // 8-byte scratch accumulator
    float*  out = (float*)d_out;

    wmse_zero_acc<<<1, 1, 0, stream>>>(acc);
    // 4096 blocks x 256 threads: ~12.5 float4 iterations/thread, enough
    // concurrency to saturate 23.3 TB/s HBM.
    wmse_reduce_kernel<<<4096, 256, 0, stream>>>(pred, targ, months, mw, aw, acc);
    wmse_finalize<<<1, 1, 0, stream>>>(acc, out);
}